// DSF_31688268710552
// MI455X (gfx1250) — compile-verified
//
#include <hip/hip_runtime.h>
#include <hip/hip_bf16.h>
#include <math.h>

// Problem constants (from reference): x:(4,64,64,64) f32, w:(2,64,3,3), b:(2,)
#define BATCH 4
#define CH    64
#define HH    64
#define WW    64
#define NPIX  (BATCH * HH * WW)     // 16384
#define KTOT  (CH * 9)              // 576
#define MAXR  100.0f

typedef __attribute__((ext_vector_type(2))) float v2f;
typedef __attribute__((ext_vector_type(8))) float v8f;
typedef __attribute__((ext_vector_type(4))) unsigned int u32x4;
typedef __attribute__((ext_vector_type(8))) int i32x8;
typedef __attribute__((ext_vector_type(4))) int i32x4;

#if __has_builtin(__builtin_amdgcn_tensor_load_to_lds) && __has_builtin(__builtin_amdgcn_s_wait_tensorcnt)
#define USE_TDM 1
#else
#define USE_TDM 0
#endif

__device__ __forceinline__ float dev_tanh(float v) {
#if __has_builtin(__builtin_amdgcn_tanhf)
    return __builtin_amdgcn_tanhf(v);          // native V_TANH_F32 on gfx1250
#elif __has_builtin(__builtin_amdgcn_tanh_f32)
    return __builtin_amdgcn_tanh_f32(v);
#else
    return tanhf(v);
#endif
}

__device__ __forceinline__ float dev_exp(float v) {
    return __builtin_amdgcn_exp2f(v * 1.4426950408889634f);
}

__device__ __forceinline__ float clampR(float v) {
    return fminf(fmaxf(v, -MAXR), MAXR);
}

// -------------------------------------------------------------------------
// Kernel 1: 3x3 conv (64ch -> 2ch) + bias + 100*tanh via V_WMMA_F32_16X16X4_F32.
// GEMM decomposed per tap: out[16pad, 16 pixels] += A_tap[16x64ch] * B_tap[64ch x16].
// Bounds handling via clamped addresses * {0,1} float mask -> all loads
// unconditional (no EXEC save/restore), no integer divides in the hot loop.
// A layout (32-bit 16x4): lanes 0-15: V0=K0,V1=K1 ; lanes 16-31: V0=K2,V1=K3.
// B layout (32-bit 4x16): V0: lanes0-15 K=0 (N=lane), lanes16-31 K=1 ;
//                         V1: lanes0-15 K=2,          lanes16-31 K=3.
// -------------------------------------------------------------------------
__global__ void __launch_bounds__(256)
conv_ab_kernel(const float* __restrict__ x,
               const float* __restrict__ cw,
               const float* __restrict__ cb,
               float* __restrict__ alpha,
               float* __restrict__ beta) {
    __shared__ float wlds[2 * KTOT];   // 1152 weight floats staged in LDS

    const int tid = threadIdx.x;
    for (int i = tid; i < 2 * KTOT; i += 256) wlds[i] = cw[i];
    __syncthreads();

    const int wave = tid >> 5;
    const int lane = tid & 31;
    const int tile = blockIdx.x * 8 + wave;     // 1024 tiles of 16 pixels
    const int p0   = tile * 16;
    const int b    = p0 >> 12;
    const int h    = (p0 >> 6) & 63;
    const int w0   = p0 & 63;                   // 0,16,32,48

    const int   half = lane >> 4;               // 0 or 1
    const int   n    = lane & 15;               // pixel within tile / A row (m)
    const float asel = (n < 2) ? 1.0f : 0.0f;   // rows >= 2 of A are zero pad
    const int   m2   = n & 1;

    v8f acc = {};
#pragma unroll 1
    for (int tap = 0; tap < 9; ++tap) {
        const int dy = tap / 3, dx = tap - dy * 3;     // once per tap only
        const int hh = h + dy - 1;                     // wave-uniform
        const int wn = w0 + n + dx - 1;                // per-lane
        const float msk =
            ((hh >= 0) & (hh < HH) & (wn >= 0) & (wn < WW)) ? 1.0f : 0.0f;
        const int hc = min(max(hh, 0), HH - 1);
        const int wc = min(max(wn, 0), WW - 1);
        const float* bp = x + (b * CH * HH * WW + hc * WW + wc);  // + c*4096
        const float* ap = wlds + (m2 * KTOT + tap);               // + c*9

#pragma unroll 4
        for (int c0 = 0; c0 < CH; c0 += 4) {
            v2f a, bb;
            a[0]  = ap[(c0 + 2 * half) * 9] * asel;
            a[1]  = ap[(c0 + 2 * half + 1) * 9] * asel;
            bb[0] = bp[(c0 + half) * (HH * WW)] * msk;
            bb[1] = bp[(c0 + 2 + half) * (HH * WW)] * msk;
            acc = __builtin_amdgcn_wmma_f32_16x16x4_f32(
                /*neg_a=*/false, a, /*neg_b=*/false, bb,
                /*c_mod=*/(short)0, acc, /*reuse_a=*/false, /*reuse_b=*/false);
        }
    }

    if (lane < 16) {   // lanes 0-15 hold M=0 (alpha) in acc[0], M=1 (beta) in acc[1]
        const int p = p0 + lane;
        alpha[p] = MAXR * dev_tanh(acc[0] + cb[0]);
        beta[p]  = MAXR * dev_tanh(acc[1] + cb[1]);
    }
}

// -------------------------------------------------------------------------
// Kernel 2: fused main op. One block = 32 consecutive pixels (same b,h row
// segment) x all 64 channels. x tile staged into LDS via the Tensor Data
// Mover (TENSOR_LOAD_TO_LDS, 2D tile: 64 rows x 32 f32, row stride 4096),
// issued by wave 0 and published with s_wait_tensorcnt + barrier.
// -------------------------------------------------------------------------
__global__ void __launch_bounds__(256)
fused_main_kernel(const float* __restrict__ x,
                  const float* __restrict__ alpha,
                  const float* __restrict__ beta,
                  float* __restrict__ out) {
    __shared__ float xs[CH][32];
    __shared__ float vs[CH][32];

    const int tid = threadIdx.x;
    const int p0  = blockIdx.x * 32;            // 512 blocks
    const int b   = p0 >> 12;
    const int h   = (p0 >> 6) & 63;
    const int w0  = p0 & 63;                    // 0 or 32

#if USE_TDM
    if (tid < 32) {                             // wave 0 issues one TDM op
        const unsigned long long ga =
            (unsigned long long)(uintptr_t)(x + (b * CH * HH * WW + h * WW + w0));
        const unsigned int ga_lo = (unsigned int)ga;
        const unsigned int ga_hi = (unsigned int)(ga >> 32);
        const unsigned int lds   = (unsigned int)(uintptr_t)&xs[0][0]; // LDS offset

        u32x4 g0;
        g0[0] = 1u;                                   // count=1, user mode
        g0[1] = lds;                                  // lds_addr
        g0[2] = ga_lo;                                // global_addr[31:0]
        g0[3] = (ga_hi & 0x01FFFFFFu) | (2u << 30);   // global_addr[56:32], type=2

        i32x8 g1;
        g1[0] = (int)(2u << 16);       // workgroup_mask=0, data_size=4B
        g1[1] = (int)(32u << 16);      // atomic_barrier_addr=0, tensor_dim0=32 (lo16)
        g1[2] = (int)(64u << 16);      // tensor_dim0 hi=0, tensor_dim1=64 (lo16)
        g1[3] = (int)(32u << 16);      // tensor_dim1 hi=0, tile_dim0=32
        g1[4] = 64;                    // tile_dim1=64, tile_dim2=0
        g1[5] = (int)(HH * WW);        // tensor_dim0_stride=4096 (lo32)
        g1[6] = 0;                     // stride hi16=0, tensor_dim1_stride lo16=0
        g1[7] = 0;                     // tensor_dim1_stride hi=0 (unused for 2D)

        i32x4 gz4 = {0, 0, 0, 0};             // groups 2/3 unused (2D tensor)
        i32x8 gz8 = {0, 0, 0, 0, 0, 0, 0, 0}; // trailing group slot unused
        // 6-arg variant (amdgpu-toolchain clang-23):
        // (u32x4, i32x8, i32x4, i32x4, i32x8, i32 cpol)
        __builtin_amdgcn_tensor_load_to_lds(g0, g1, gz4, gz4, gz8, 0);
        __builtin_amdgcn_s_wait_tensorcnt(0);
    }
#else
    for (int idx = tid; idx < CH * 32; idx += 256) {
        int c = idx >> 5, ww = idx & 31;
        xs[c][ww] = x[b * (CH * HH * WW) + c * (HH * WW) + h * WW + w0 + ww];
    }
#endif
    __syncthreads();

    const int wl = tid & 31;                    // pixel lane
    const int jg = tid >> 5;                    // j-group (8 groups of 8)
    const float al = alpha[p0 + wl];
    const float be = beta[p0 + wl];

#pragma unroll
    for (int jj = 0; jj < 8; ++jj) {
        const int j = jg * 8 + jj;
        const float xj = xs[j][wl];
        float acc = 0.0f;
#pragma unroll 8
        for (int i = 0; i < CH; ++i) {
            float d  = fmaf(be, xs[i][wl], -xj);         // beta*x_i - x_j
            float td = dev_tanh(clampR(al * d));         // tanh(clip(alpha*d))
            acc = fmaf(d, td, acc);
        }
        float tmp = acc * (1.0f / 64.0f);                // mean over i
        vs[j][wl] = clampR(-tmp * al);                   // softmax logits
    }
    __syncthreads();

    if (tid < 32) {                                      // one lane per pixel
        float mx = -3.0e38f;
#pragma unroll 8
        for (int j = 0; j < CH; ++j) mx = fmaxf(mx, vs[j][tid]);
        float se = 0.0f, sx = 0.0f;
#pragma unroll 8
        for (int j = 0; j < CH; ++j) {
            float e = dev_exp(vs[j][tid] - mx);
            se += e;
            sx = fmaf(e, xs[j][tid], sx);
        }
        out[p0 + tid] = sx / se;                         // sum_j x_j * softmax_j
    }
}

extern "C" void kernel_launch(void* const* d_in, const int* in_sizes, int n_in,
                              void* d_out, int out_size, void* d_ws, size_t ws_size,
                              hipStream_t stream) {
    (void)in_sizes; (void)n_in; (void)out_size; (void)ws_size;
    const float* x  = (const float*)d_in[0];   // (4,64,64,64)
    const float* cw = (const float*)d_in[1];   // (2,64,3,3)
    const float* cb = (const float*)d_in[2];   // (2,)
    float* out   = (float*)d_out;              // (4,1,64,64) -> 16384 floats
    float* alpha = (float*)d_ws;               // 16384 floats
    float* beta  = alpha + NPIX;               // 16384 floats

    conv_ab_kernel<<<NPIX / (16 * 8), 256, 0, stream>>>(x, cw, cb, alpha, beta);
    fused_main_kernel<<<NPIX / 32, 256, 0, stream>>>(x, alpha, beta, out);
}